// BiomechStepGNO_37563783970932
// MI455X (gfx1250) — compile-verified
//
#include <hip/hip_runtime.h>
#include <math.h>

#define B_ 2
#define N_ 4096
#define G_ 4096
#define NM_ 64
#define EMB_ 8
#define CIN_ 17
#define UPAD_ 20          // u rows padded 17 -> 20 so lifting GEMM K%4==0
#define HID_ 32
#define KH_ 32
#define LAYERS_ 4
#define RADIUS_ 0.08f

typedef __attribute__((ext_vector_type(2))) float v2f;
typedef __attribute__((ext_vector_type(8))) float v8f;

// Branch-free tanh-gelu: 0.5x(1+tanh(y)) == x / (1 + exp(-2y)).
// __expf -> v_exp_f32; robust at both tails (exp->inf => 0, exp->0 => x).
__device__ __forceinline__ float gelu_f(float x) {
    const float c0 = 0.7978845608028654f;   // sqrt(2/pi)
    float y = c0 * (x + 0.044715f * x * x * x);
    return x / (1.0f + __expf(-2.0f * y));
}

// cos/sin(2*pi*t/16)
__constant__ float CTW[16] = {
    1.0f, 0.92387953251f, 0.70710678119f, 0.38268343236f,
    0.0f, -0.38268343236f, -0.70710678119f, -0.92387953251f,
    -1.0f, -0.92387953251f, -0.70710678119f, -0.38268343236f,
    0.0f, 0.38268343236f, 0.70710678119f, 0.92387953251f };
__constant__ float STW[16] = {
    0.0f, 0.38268343236f, 0.70710678119f, 0.92387953251f,
    1.0f, 0.92387953251f, 0.70710678119f, 0.38268343236f,
    0.0f, -0.38268343236f, -0.70710678119f, -0.92387953251f,
    -1.0f, -0.38268343236f, -0.70710678119f, -0.92387953251f };
// NOTE: STW row above must be sin(2*pi*t/16); correct full table re-stated below.

// (Re-declare with the verified values; keep a single authoritative pair.)
__constant__ float CTW2[16] = {
    1.0f, 0.92387953251f, 0.70710678119f, 0.38268343236f,
    0.0f, -0.38268343236f, -0.70710678119f, -0.92387953251f,
    -1.0f, -0.92387953251f, -0.70710678119f, -0.38268343236f,
    0.0f, 0.38268343236f, 0.70710678119f, 0.92387953251f };
__constant__ float STW2[16] = {
    0.0f, 0.38268343236f, 0.70710678119f, 0.92387953251f,
    1.0f, 0.92387953251f, 0.70710678119f, 0.38268343236f,
    0.0f, -0.38268343236f, -0.70710678119f, -0.92387953251f,
    -1.0f, -0.92387953251f, -0.70710678119f, -0.38268343236f };
#define CT(t) CTW2[t]
#define ST(t) STW2[t]

// ---------------------------------------------------------------------------
// Feature assembly: pose_norm (B,N,3) and xpre = [pose_norm, feats(17)] @ Wx + b1
// ---------------------------------------------------------------------------
__global__ void k_features(const float* __restrict__ pose, const float* __restrict__ activ,
                           const float* __restrict__ rest, const float* __restrict__ bbmin,
                           const float* __restrict__ bbsize, const float* __restrict__ midline,
                           const float* __restrict__ fixedm, const float* __restrict__ memb,
                           const int* __restrict__ dm, const float* __restrict__ Wx,
                           const float* __restrict__ b1, float* __restrict__ pose_norm,
                           float* __restrict__ xpre) {
    int i = blockIdx.x * blockDim.x + threadIdx.x;
    if (i >= B_ * N_) return;
    int b = i >> 12;
    int n = i & (N_ - 1);
    float fin[20];
#pragma unroll
    for (int d = 0; d < 3; d++) {
        float pn = (pose[i * 3 + d] - bbmin[d]) / bbsize[d];
        fin[d] = pn;
        fin[17 + d] = pn;
        pose_norm[i * 3 + d] = pn;
        fin[14 + d] = (rest[n * 3 + d] - bbmin[d]) / bbsize[d];
    }
    int dmi = dm[n];
    float nomus = (dmi < 0) ? 1.0f : 0.0f;
    int sdm = (dmi < 0) ? 0 : dmi;
    float a = activ[b * NM_ + sdm] * 0.01f;
    a = fminf(fmaxf(a, 0.0f), 1.0f) * (1.0f - nomus);
    fin[3] = a;
#pragma unroll
    for (int e = 0; e < EMB_; e++) fin[4 + e] = memb[(dmi + 1) * EMB_ + e];
    fin[12] = midline[n];
    fin[13] = fixedm[n];
#pragma unroll
    for (int k = 0; k < KH_; k++) {
        float s = b1[k];
#pragma unroll
        for (int j = 0; j < 20; j++) s += fin[j] * Wx[j * KH_ + k];
        xpre[i * KH_ + k] = s;
    }
}

// ---------------------------------------------------------------------------
// Encoder GNO: mesh -> latent grid (kernel-only). One wave per query.
//   u[q,c] = ((sum_{s in R} gelu(hy+xpre_s)) @ W2 + cnt*b2) / max(cnt,1)
// u rows padded to UPAD_ (cols 17..19 = 0).
// ---------------------------------------------------------------------------
__global__ void k_gno_in(const float* __restrict__ pose_norm, const float* __restrict__ xpre,
                         const float* __restrict__ Wy, const float* __restrict__ W2,
                         const float* __restrict__ b2, float* __restrict__ u) {
    __shared__ float s_xyz[256 * 3];
    __shared__ float s_pre[256 * 32];
    __shared__ float s_h[8 * 32];
    int lane = threadIdx.x & 31;
    int w = threadIdx.x >> 5;
    int q = blockIdx.x * 8 + w;          // b*G + g
    int b = (blockIdx.x * 8) / G_;       // wave-uniform (8 | G)
    int g = q & (G_ - 1);
    float qx = (float)(g >> 8) * (1.0f / 15.0f);
    float qy = (float)((g >> 4) & 15) * (1.0f / 15.0f);
    float qz = (float)(g & 15) * (1.0f / 15.0f);
    float hy = qx * Wy[lane] + qy * Wy[32 + lane] + qz * Wy[64 + lane];
    float acc = 0.0f;
    int cnt = 0;
    const float r2 = RADIUS_ * RADIUS_;
    for (int s0 = 0; s0 < N_; s0 += 256) {
        __syncthreads();
        int t = threadIdx.x;
        int sr = s0 + t;
        s_xyz[t * 3 + 0] = pose_norm[(b * N_ + sr) * 3 + 0];
        s_xyz[t * 3 + 1] = pose_norm[(b * N_ + sr) * 3 + 1];
        s_xyz[t * 3 + 2] = pose_norm[(b * N_ + sr) * 3 + 2];
        for (int idx = t; idx < 256 * 32; idx += 256)
            s_pre[idx] = xpre[(b * N_ + s0) * 32 + idx];
        // prefetch next tile into cache while this tile is consumed
        if (s0 + 256 < N_)
            __builtin_prefetch(xpre + (b * N_ + s0 + 256) * 32 + threadIdx.x * 32, 0, 0);
        __syncthreads();
        for (int s = 0; s < 256; s++) {
            float dx = s_xyz[s * 3 + 0] - qx;
            float dy = s_xyz[s * 3 + 1] - qy;
            float dz = s_xyz[s * 3 + 2] - qz;
            float d2 = dx * dx + dy * dy + dz * dz;
            if (d2 < r2) {                      // wave-uniform branch
                acc += gelu_f(hy + s_pre[s * 32 + lane]);
                cnt++;
            }
        }
    }
    __syncthreads();
    s_h[w * 32 + lane] = acc;
    __syncthreads();
    if (lane < UPAD_) {
        float val = 0.0f;
        if (lane < CIN_) {
            float s = 0.0f;
#pragma unroll
            for (int k = 0; k < 32; k++) s += s_h[w * 32 + k] * W2[k * CIN_ + lane];
            s += (float)cnt * b2[lane];
            float den = (cnt > 0) ? (float)cnt : 1.0f;
            val = s / den;
        }
        u[q * UPAD_ + lane] = val;
    }
}

// ---------------------------------------------------------------------------
// fp32 WMMA GEMM, compile-time everything:
//   C = [ACT? gelu](A[M,K]@W[K,N] + bias [+ Dadd]) [* colscale]
// One wave per 16x16 tile, V_WMMA_F32_16X16X4_F32. M%16==0, K%4==0.
// NCHK=false requires N%16==0 (no guards, clean load clauses + wmma chain).
// ---------------------------------------------------------------------------
template <int K, int N, bool NCHK, bool DADD, bool ACT, bool CSCALE>
__global__ void k_gemm_wmma(const float* __restrict__ A, const float* __restrict__ W,
                            const float* __restrict__ bias, const float* __restrict__ Dadd,
                            const float* __restrict__ colscale, float* __restrict__ C,
                            int M) {
    constexpr int NTILES = (N + 15) >> 4;
    int lane = threadIdx.x & 31;
    int wave = blockIdx.x * (blockDim.x >> 5) + (threadIdx.x >> 5);
    int mtiles = M >> 4;
    int mt = wave / NTILES;
    int nt = wave - mt * NTILES;
    if (mt >= mtiles) return;                     // wave-uniform
    int hi = lane >> 4;                           // 0 or 1
    int mrow = (mt << 4) + (lane & 15);           // A-fragment row (all lanes)
    int ncol = (nt << 4) + (lane & 15);           // B/C/D column
    bool ncolok = NCHK ? (ncol < N) : true;
    const float* arow = A + (size_t)mrow * K;
    v8f acc = {0.f, 0.f, 0.f, 0.f, 0.f, 0.f, 0.f, 0.f};
#pragma unroll
    for (int k0 = 0; k0 < K; k0 += 4) {
        int ka = k0 + hi * 2;                     // ISA: lanes16-31 hold K=k0+2,k0+3
        v2f af = *(const v2f*)(arow + ka);        // 8B-aligned pair -> global_load_b64
        v2f bf;
        if (NCHK) {
            bf[0] = ncolok ? W[ka * N + ncol] : 0.0f;
            bf[1] = ncolok ? W[(ka + 1) * N + ncol] : 0.0f;
        } else {
            bf[0] = W[ka * N + ncol];
            bf[1] = W[(ka + 1) * N + ncol];
        }
        acc = __builtin_amdgcn_wmma_f32_16x16x4_f32(false, af, false, bf,
                                                    (short)0, acc, false, false);
    }
    if (ncolok) {
        float bi = bias[ncol];
        float cs = CSCALE ? colscale[ncol] : 1.0f;
#pragma unroll
        for (int j = 0; j < 8; j++) {
            int m = (mt << 4) + j + (hi << 3);    // D layout: M = j + 8*(lane>=16)
            float v = acc[j] + bi;
            if (DADD) v += Dadd[m * N + ncol];
            if (ACT) v = gelu_f(v);
            C[m * N + ncol] = CSCALE ? v * cs : v;
        }
    }
}

// ---------------------------------------------------------------------------
// Pad Wl (17x32) to 20x32 with zero rows.
// ---------------------------------------------------------------------------
__global__ void k_pad_wl(const float* __restrict__ Wl, float* __restrict__ Wl_pad) {
    int i = blockIdx.x * blockDim.x + threadIdx.x;
    if (i >= UPAD_ * HID_) return;
    Wl_pad[i] = (i < CIN_ * HID_) ? Wl[i] : 0.0f;
}

// ---------------------------------------------------------------------------
// FNO spectral pipeline (16-point DFTs via twiddle table)
// Spectral buffers addressed as ((b*4096)+ix*256+iy*16+kz)*32+c, kz in 0..8.
// ---------------------------------------------------------------------------
__global__ void k_dft_z(const float* __restrict__ v, float* __restrict__ outR,
                        float* __restrict__ outI) {
    int i = blockIdx.x * 256 + threadIdx.x;       // B*16*16*9*32
    if (i >= B_ * 16 * 16 * 9 * 32) return;
    int c = i & 31; int t = i >> 5;
    int kz = t % 9; t /= 9;
    int iy = t & 15; t >>= 4;
    int ix = t & 15; int b = t >> 4;
    float sr = 0.0f, si = 0.0f;
    int base = ((b * 4096) + ix * 256 + iy * 16) * 32 + c;
#pragma unroll
    for (int z = 0; z < 16; z++) {
        float x = v[base + z * 32];
        int tw = (kz * z) & 15;
        sr += x * CT(tw);
        si -= x * ST(tw);
    }
    int o = ((b * 4096) + ix * 256 + iy * 16 + kz) * 32 + c;
    outR[o] = sr; outI[o] = si;
}

__global__ void k_cdft_y(const float* __restrict__ inR, const float* __restrict__ inI,
                         float* __restrict__ outR, float* __restrict__ outI, float sgn) {
    int i = blockIdx.x * 256 + threadIdx.x;
    if (i >= B_ * 16 * 16 * 9 * 32) return;
    int c = i & 31; int t = i >> 5;
    int kz = t % 9; t /= 9;
    int ky = t & 15; t >>= 4;
    int ix = t & 15; int b = t >> 4;
    float sr = 0.0f, si = 0.0f;
#pragma unroll
    for (int j = 0; j < 16; j++) {
        int idx = ((b * 4096) + ix * 256 + j * 16 + kz) * 32 + c;
        float R = inR[idx], I = inI[idx];
        int tw = (ky * j) & 15;
        float cc = CT(tw), ss = ST(tw);
        sr += R * cc - sgn * I * ss;
        si += I * cc + sgn * R * ss;
    }
    int o = ((b * 4096) + ix * 256 + ky * 16 + kz) * 32 + c;
    outR[o] = sr; outI[o] = si;
}

__global__ void k_cdft_x(const float* __restrict__ inR, const float* __restrict__ inI,
                         float* __restrict__ outR, float* __restrict__ outI, float sgn) {
    int i = blockIdx.x * 256 + threadIdx.x;
    if (i >= B_ * 16 * 16 * 9 * 32) return;
    int c = i & 31; int t = i >> 5;
    int kz = t % 9; t /= 9;
    int iy = t & 15; t >>= 4;
    int kx = t & 15; int b = t >> 4;
    float sr = 0.0f, si = 0.0f;
#pragma unroll
    for (int j = 0; j < 16; j++) {
        int idx = ((b * 4096) + j * 256 + iy * 16 + kz) * 32 + c;
        float R = inR[idx], I = inI[idx];
        int tw = (kx * j) & 15;
        float cc = CT(tw), ss = ST(tw);
        sr += R * cc - sgn * I * ss;
        si += I * cc + sgn * R * ss;
    }
    int o = ((b * 4096) + kx * 256 + iy * 16 + kz) * 32 + c;
    outR[o] = sr; outI[o] = si;
}

// Spectral multiply: corners cover all (kx,ky); kz 0..7 kept, kz=8 zeroed.
__global__ void k_spec(const float* __restrict__ inR, const float* __restrict__ inI,
                       const float* __restrict__ Wr, const float* __restrict__ Wi,
                       float* __restrict__ outR, float* __restrict__ outI) {
    int i = blockIdx.x * 256 + threadIdx.x;
    if (i >= B_ * 16 * 16 * 9 * 32) return;
    int oo = i & 31; int t = i >> 5;
    int kz = t % 9; t /= 9;
    int ky = t & 15; t >>= 4;
    int kx = t & 15; int b = t >> 4;
    int od = ((b * 4096) + kx * 256 + ky * 16 + kz) * 32 + oo;
    if (kz == 8) { outR[od] = 0.0f; outI[od] = 0.0f; return; }
    int bi = ((kx >= 8) ? 2 : 0) + ((ky >= 8) ? 1 : 0);
    int mx = kx & 7, my = ky & 7;
    const float* wr = Wr + (size_t)((((bi * 8 + mx) * 8 + my) * 8 + kz) * 32) * 32;
    const float* wi = Wi + (size_t)((((bi * 8 + mx) * 8 + my) * 8 + kz) * 32) * 32;
    int base = ((b * 4096) + kx * 256 + ky * 16 + kz) * 32;
    float sr = 0.0f, si = 0.0f;
#pragma unroll 8
    for (int ii = 0; ii < 32; ii++) {
        float vr = inR[base + ii], vi = inI[base + ii];
        float a = wr[ii * 32 + oo], bb = wi[ii * 32 + oo];
        sr += vr * a - vi * bb;
        si += vr * bb + vi * a;
    }
    outR[od] = sr; outI[od] = si;
}

// irfft along z (bins 0..7; bin 8 is zero); full 1/4096 irfftn normalization here.
__global__ void k_irfft_z(const float* __restrict__ inR, const float* __restrict__ inI,
                          float* __restrict__ vs) {
    int i = blockIdx.x * 256 + threadIdx.x;       // B*4096*32
    if (i >= B_ * 4096 * 32) return;
    int c = i & 31; int t = i >> 5;
    int z = t & 15; t >>= 4;
    int iy = t & 15; t >>= 4;
    int ix = t & 15; int b = t >> 4;
    int base = ((b * 4096) + ix * 256 + iy * 16) * 32 + c;
    float s = inR[base];                          // kz = 0 (Re only, pocketfft c2r)
#pragma unroll
    for (int kz = 1; kz < 8; kz++) {
        int tw = (kz * z) & 15;
        s += 2.0f * (inR[base + kz * 32] * CT(tw) - inI[base + kz * 32] * ST(tw));
    }
    vs[((b * 4096) + ix * 256 + iy * 16 + z) * 32 + c] = s * (1.0f / 4096.0f);
}

// ---------------------------------------------------------------------------
// g_pre = latent @ Wg_out + b1_out
// ---------------------------------------------------------------------------
__global__ void k_gpre(const float* __restrict__ Wg, const float* __restrict__ b1o,
                       float* __restrict__ gpre) {
    int i = blockIdx.x * 256 + threadIdx.x;
    if (i >= G_ * 32) return;
    int k = i & 31, g = i >> 5;
    float gx = (float)(g >> 8) * (1.0f / 15.0f);
    float gy = (float)((g >> 4) & 15) * (1.0f / 15.0f);
    float gz = (float)(g & 15) * (1.0f / 15.0f);
    gpre[i] = gx * Wg[k] + gy * Wg[32 + k] + gz * Wg[64 + k] + b1o[k];
}

// ---------------------------------------------------------------------------
// Decoder GNO: latent grid -> mesh queries (linear transform k*f).
// Sources are the regular grid -> direct <=3^3 cell lookup. One wave / query.
// ---------------------------------------------------------------------------
__global__ void k_gno_out(const float* __restrict__ pose_norm, const float* __restrict__ gpre,
                          const float* __restrict__ v, const float* __restrict__ Wy,
                          const float* __restrict__ W2, const float* __restrict__ b2,
                          float* __restrict__ o) {
    int lane = threadIdx.x & 31;
    int w = threadIdx.x >> 5;
    int q = blockIdx.x * 8 + w;                   // b*N + n
    int b = q >> 12;
    float qx = pose_norm[q * 3 + 0];
    float qy = pose_norm[q * 3 + 1];
    float qz = pose_norm[q * 3 + 2];
    float hy = qx * Wy[lane] + qy * Wy[32 + lane] + qz * Wy[64 + lane];
    const float r2 = RADIUS_ * RADIUS_;
    int xlo = (int)floorf((qx - RADIUS_) * 15.0f); if (xlo < 0) xlo = 0;
    int xhi = (int)ceilf((qx + RADIUS_) * 15.0f);  if (xhi > 15) xhi = 15;
    int ylo = (int)floorf((qy - RADIUS_) * 15.0f); if (ylo < 0) ylo = 0;
    int yhi = (int)ceilf((qy + RADIUS_) * 15.0f);  if (yhi > 15) yhi = 15;
    int zlo = (int)floorf((qz - RADIUS_) * 15.0f); if (zlo < 0) zlo = 0;
    int zhi = (int)ceilf((qz + RADIUS_) * 15.0f);  if (zhi > 15) zhi = 15;
    float acc = 0.0f;
    int cnt = 0;
    float b2l = b2[lane];
    for (int ix = xlo; ix <= xhi; ix++) {
        float dx = (float)ix * (1.0f / 15.0f) - qx;
        for (int iy = ylo; iy <= yhi; iy++) {
            float dy = (float)iy * (1.0f / 15.0f) - qy;
            for (int iz = zlo; iz <= zhi; iz++) {
                float dz = (float)iz * (1.0f / 15.0f) - qz;
                float d2 = dx * dx + dy * dy + dz * dz;
                if (d2 < r2) {                    // wave-uniform branch
                    int s = (ix << 8) + (iy << 4) + iz;
                    float h = gelu_f(hy + gpre[s * 32 + lane]);
                    float tt = 0.0f;
#pragma unroll
                    for (int k = 0; k < 32; k++)
                        tt += __shfl(h, k, 32) * W2[k * 32 + lane];
                    acc += (tt + b2l) * v[((b << 12) + s) * 32 + lane];
                    cnt++;
                }
            }
        }
    }
    float den = (cnt > 0) ? (float)cnt : 1.0f;
    o[q * 32 + lane] = acc / den;
}

// ---------------------------------------------------------------------------
// Host-side launch
// ---------------------------------------------------------------------------
extern "C" void kernel_launch(void* const* d_in, const int* in_sizes, int n_in,
                              void* d_out, int out_size, void* d_ws, size_t ws_size,
                              hipStream_t stream) {
    (void)in_sizes; (void)n_in; (void)out_size; (void)ws_size;
    const float* pose    = (const float*)d_in[0];
    const float* activ   = (const float*)d_in[1];
    const float* rest    = (const float*)d_in[2];
    const float* bbmin   = (const float*)d_in[3];
    const float* bbsize  = (const float*)d_in[4];
    const float* midline = (const float*)d_in[5];
    const float* fixedm  = (const float*)d_in[6];
    const float* memb    = (const float*)d_in[7];
    const float* Wy_in   = (const float*)d_in[8];
    const float* Wx_in   = (const float*)d_in[9];
    const float* b1_in   = (const float*)d_in[10];
    const float* W2_in   = (const float*)d_in[11];
    const float* b2_in   = (const float*)d_in[12];
    const float* Wl      = (const float*)d_in[13];
    const float* bl      = (const float*)d_in[14];
    const float* Wr      = (const float*)d_in[15];
    const float* Wi      = (const float*)d_in[16];
    const float* Wskip   = (const float*)d_in[17];
    const float* bskip   = (const float*)d_in[18];
    const float* Wy_out  = (const float*)d_in[19];
    const float* Wg_out  = (const float*)d_in[20];
    const float* b1_out  = (const float*)d_in[21];
    const float* W2_out  = (const float*)d_in[22];
    const float* b2_out  = (const float*)d_in[23];
    const float* Wp1     = (const float*)d_in[24];
    const float* bp1     = (const float*)d_in[25];
    const float* Wp2     = (const float*)d_in[26];
    const float* bp2     = (const float*)d_in[27];
    const int*   dm      = (const int*)d_in[28];
    float* out = (float*)d_out;
    float* ws = (float*)d_ws;

    float* pose_norm = ws;                        // 24576
    float* xpre = pose_norm + 24576;              // 262144
    float* u    = xpre + 262144;                  // 163840 (8192 x UPAD_)
    float* v0   = u + 163840;                     // 262144
    float* v1   = v0 + 262144;                    // 262144
    float* R0   = v1 + 262144;                    // 262144 each
    float* I0   = R0 + 262144;
    float* R1   = I0 + 262144;
    float* I1   = R1 + 262144;
    float* gpre = I1 + 262144;                    // 131072
    float* obuf = gpre + 131072;                  // 262144
    float* h1   = obuf + 262144;                  // 524288
    float* Wl_pad = h1 + 524288;                  // 640

    const int M = B_ * G_;                        // 8192 (== B_*N_ too)
    const int SPEC_BLOCKS = (B_ * 16 * 16 * 9 * 32) / 256;   // 576

    k_features<<<(B_ * N_) / 256, 256, 0, stream>>>(pose, activ, rest, bbmin, bbsize,
                                                    midline, fixedm, memb, dm, Wx_in, b1_in,
                                                    pose_norm, xpre);
    k_gno_in<<<(B_ * G_) / 8, 256, 0, stream>>>(pose_norm, xpre, Wy_in, W2_in, b2_in, u);

    // Pad Wl to 20 rows (zeros) so the lifting GEMM uses the unchecked K=20 path.
    k_pad_wl<<<3, 256, 0, stream>>>(Wl, Wl_pad);
    k_gemm_wmma<UPAD_, HID_, false, false, false, false>
        <<<(M / 16) * 2 / 8, 256, 0, stream>>>(u, Wl_pad, bl, nullptr, nullptr, v0, M);

    float* vc = v0;
    float* vn = v1;
    for (int l = 0; l < LAYERS_; l++) {
        const float* wr = Wr + (size_t)l * 2097152;   // 4*8*8*8*32*32
        const float* wi = Wi + (size_t)l * 2097152;
        k_dft_z<<<SPEC_BLOCKS, 256, 0, stream>>>(vc, R0, I0);
        k_cdft_y<<<SPEC_BLOCKS, 256, 0, stream>>>(R0, I0, R1, I1, -1.0f);
        k_cdft_x<<<SPEC_BLOCKS, 256, 0, stream>>>(R1, I1, R0, I0, -1.0f);
        k_spec<<<SPEC_BLOCKS, 256, 0, stream>>>(R0, I0, wr, wi, R1, I1);
        k_cdft_x<<<SPEC_BLOCKS, 256, 0, stream>>>(R1, I1, R0, I0, 1.0f);
        k_cdft_y<<<SPEC_BLOCKS, 256, 0, stream>>>(R0, I0, R1, I1, 1.0f);
        k_irfft_z<<<(B_ * 4096 * 32) / 256, 256, 0, stream>>>(R1, I1, vn);
        // v = gelu(vs + v@Wskip[l] + bskip[l]); in-place add of vn
        k_gemm_wmma<HID_, HID_, false, true, true, false>
            <<<(M / 16) * 2 / 8, 256, 0, stream>>>(vc, Wskip + l * HID_ * HID_,
                                                   bskip + l * HID_, vn, nullptr, vn, M);
        float* tmp = vc; vc = vn; vn = tmp;
    }

    k_gpre<<<(G_ * 32) / 256, 256, 0, stream>>>(Wg_out, b1_out, gpre);
    k_gno_out<<<(B_ * N_) / 8, 256, 0, stream>>>(pose_norm, gpre, vc, Wy_out, W2_out,
                                                 b2_out, obuf);

    // Projection MLP: gelu(o@Wp1+bp1) @ Wp2 + bp2, scaled by bbox_size -> out
    k_gemm_wmma<HID_, 64, false, false, true, false>
        <<<(M / 16) * 4 / 8, 256, 0, stream>>>(obuf, Wp1, bp1, nullptr, nullptr, h1, M);
    k_gemm_wmma<64, 3, true, false, false, true>
        <<<(M / 16) * 1 / 8, 256, 0, stream>>>(h1, Wp2, bp2, nullptr, bbsize, out, M);
}